// NodeGCN_39350490365953
// MI455X (gfx1250) — compile-verified
//
#include <hip/hip_runtime.h>

typedef __attribute__((ext_vector_type(2))) float v2f;
typedef __attribute__((ext_vector_type(8))) float v8f;

// ---------------- degree / CSR construction ----------------

__global__ void k_deg_init(int* __restrict__ deg, int n) {
  int i = blockIdx.x * blockDim.x + threadIdx.x;
  if (i < n) deg[i] = 1;                      // self-loop contributes 1
}

__global__ void k_deg_count(const int* __restrict__ dst, int e, int* __restrict__ deg) {
  int i = blockIdx.x * blockDim.x + threadIdx.x;
  if (i < e) atomicAdd(&deg[dst[i]], 1);
}

// per-block exclusive scan (256 elements), emits block totals
__global__ void k_scan1(const int* __restrict__ deg, int n,
                        int* __restrict__ rowptr, int* __restrict__ bsums) {
  __shared__ int sh[256];
  int tid = threadIdx.x;
  int i = blockIdx.x * 256 + tid;
  int v = (i < n) ? deg[i] : 0;
  sh[tid] = v;
  __syncthreads();
  for (int off = 1; off < 256; off <<= 1) {
    int t = (tid >= off) ? sh[tid - off] : 0;
    __syncthreads();
    sh[tid] += t;
    __syncthreads();
  }
  if (i < n) rowptr[i] = sh[tid] - v;         // local exclusive
  if (tid == 255) bsums[blockIdx.x] = sh[tid];
}

// single-block exclusive scan of block sums (nb <= 512)
__global__ void k_scan2(int* __restrict__ bsums, int nb) {
  __shared__ int sh[512];
  int tid = threadIdx.x;
  int v = (tid < nb) ? bsums[tid] : 0;
  sh[tid] = v;
  __syncthreads();
  for (int off = 1; off < 512; off <<= 1) {
    int t = (tid >= off) ? sh[tid - off] : 0;
    __syncthreads();
    sh[tid] += t;
    __syncthreads();
  }
  if (tid < nb) bsums[tid] = sh[tid] - v;
}

__global__ void k_finish(int* __restrict__ rowptr, int* __restrict__ cursor,
                         const int* __restrict__ bsums, const int* __restrict__ deg,
                         float* __restrict__ dinv, int n, int etotal) {
  int i = blockIdx.x * blockDim.x + threadIdx.x;
  if (i < n) {
    int r = rowptr[i] + bsums[i >> 8];
    rowptr[i] = r;
    cursor[i] = r;
    dinv[i]   = rsqrtf((float)deg[i]);        // deg >= 1 always
  }
  if (i == 0) rowptr[n] = etotal;
}

__global__ void k_fill(const int* __restrict__ src, const int* __restrict__ dst,
                       int e, int n, int* __restrict__ cursor, int* __restrict__ csrc) {
  int i = blockIdx.x * blockDim.x + threadIdx.x;
  if (i >= e + n) return;
  int s, d;
  if (i < e) { s = src[i]; d = dst[i]; }
  else       { s = d = i - e; }               // self loop
  int pos = atomicAdd(&cursor[d], 1);
  csrc[pos] = s;
}

// zero-pad W2[K,40] -> Wpad[K,48]
__global__ void k_pad_w(const float* __restrict__ W, float* __restrict__ Wp,
                        int K, int Nin, int Npad) {
  int i = blockIdx.x * blockDim.x + threadIdx.x;
  if (i >= K * Npad) return;
  int r = i / Npad, c = i - r * Npad;
  Wp[i] = (c < Nin) ? W[r * Nin + c] : 0.0f;
}

// ---------------- fp32 WMMA GEMM: C[M,NV] = A[M,K](LDA) @ B[K,NV] ----------------
// All dims compile-time. Each wave computes a (16*MT) x (16*NT) register tile via
// MT*NT accumulators of V_WMMA_F32_16X16X4_F32. Grids sized exactly: EXEC all-1s.
// All addresses are affine GEPs rooted at the kernel args -> global_load codegen.

template <int WPB, int K, int LDA, int NV, int MT, int NT>
__global__ void k_gemm(const float* __restrict__ A, const float* __restrict__ B,
                       float* __restrict__ C) {
  constexpr int ntSuperN = (NV / 16) / NT;
  int wave = blockIdx.x * WPB + (threadIdx.x >> 5);
  int lane = threadIdx.x & 31;
  int sM = wave / ntSuperN, sN = wave % ntSuperN;
  int m0 = sM * (16 * MT);
  int n0 = sN * (16 * NT);
  int lr = lane & 15;          // row (A/C) or column (B/C) within tile
  int lh = lane >> 4;          // half-wave selector

  const size_t aOff = (size_t)(m0 + lr) * LDA + 2 * lh;       // K = k + 2*lh + {0,1}
  const size_t bOff = (size_t)(2 * lh) * NV + (n0 + lr);

  v8f c[MT][NT];
#pragma unroll
  for (int mi = 0; mi < MT; mi++)
#pragma unroll
    for (int ni = 0; ni < NT; ni++) c[mi][ni] = (v8f){};

  for (int k = 0; k < K; k += 4) {
    v2f a[MT], b[NT];
#pragma unroll
    for (int mi = 0; mi < MT; mi++)
      a[mi] = *(const v2f*)(A + aOff + (size_t)(16 * mi) * LDA + k);
#pragma unroll
    for (int ni = 0; ni < NT; ni++) {
      b[ni].x = B[bOff + 16 * ni + (size_t)k * NV];           // K = k + 2*lh
      b[ni].y = B[bOff + 16 * ni + (size_t)(k + 1) * NV];     // K = k + 2*lh + 1
    }
#pragma unroll
    for (int mi = 0; mi < MT; mi++)
#pragma unroll
      for (int ni = 0; ni < NT; ni++)
        c[mi][ni] = __builtin_amdgcn_wmma_f32_16x16x4_f32(
            false, a[mi], false, b[ni], (short)0, c[mi][ni], false, false);
  }

#pragma unroll
  for (int mi = 0; mi < MT; mi++) {
#pragma unroll
    for (int ni = 0; ni < NT; ni++) {
      size_t cOff = (size_t)(m0 + 16 * mi + 8 * lh) * NV + (n0 + 16 * ni + lr);
#pragma unroll
      for (int v = 0; v < 8; v++)
        C[cOff + (size_t)v * NV] = c[mi][ni][v];              // M = v + 8*lh
    }
  }
}

// ---------------- CSR aggregation: out[d] = dinv[d] * sum_s dinv[s]*H[s] + b ----------------
// Block per destination node. Edge indices + dinv[src] staged cooperatively in LDS,
// feature lanes then read them via broadcast ds_load; only the 512B row gathers hit L2.

__global__ void k_aggregate(const float* __restrict__ H, int ldH,
                            const float* __restrict__ dinv,
                            const int* __restrict__ rowptr, const int* __restrict__ csrc,
                            const float* __restrict__ bias, float* __restrict__ out,
                            int F, int doRelu) {
  __shared__ int   sS[128];
  __shared__ float sW[128];
  int d  = blockIdx.x;
  int t  = threadIdx.x;
  int nt = blockDim.x;
  int beg = rowptr[d], end = rowptr[d + 1];
  float acc = 0.0f;
  for (int base = beg; base < end; base += nt) {
    int cnt = min(nt, end - base);
    __syncthreads();                          // previous chunk fully consumed
    if (t < cnt) {
      int s = csrc[base + t];
      sS[t] = s;
      sW[t] = dinv[s];
    }
    __syncthreads();
    if (t < F) {
      for (int j = 0; j < cnt; j++)
        acc = fmaf(sW[j], H[(size_t)sS[j] * ldH + t], acc);
    }
  }
  if (t < F) {
    float r = fmaf(dinv[d], acc, bias[t]);
    if (doRelu) r = fmaxf(r, 0.0f);
    out[(size_t)d * F + t] = r;
  }
}

// ---------------- host launcher ----------------

extern "C" void kernel_launch(void* const* d_in, const int* in_sizes, int n_in,
                              void* d_out, int out_size, void* d_ws, size_t ws_size,
                              hipStream_t stream) {
  (void)n_in; (void)out_size; (void)ws_size;
  const float* x  = (const float*)d_in[0];
  const int*   ei = (const int*)d_in[1];
  const float* W0 = (const float*)d_in[2];
  const float* b0 = (const float*)d_in[3];
  const float* W1 = (const float*)d_in[4];
  const float* b1 = (const float*)d_in[5];
  const float* W2 = (const float*)d_in[6];
  const float* b2 = (const float*)d_in[7];

  constexpr int Fin = 500, Fh = 128, Fc = 40, Fcp = 48;
  const int N = in_sizes[0] / Fin;            // 100000 (multiple of 32)
  const int E = in_sizes[1] / 2;              // 1600000
  const int* src = ei;
  const int* dst = ei + E;
  const int Etot = E + N;

  char* w = (char*)d_ws;
  auto carve = [&](size_t bytes) -> void* {
    void* p = (void*)w;
    w += (bytes + 255) & ~(size_t)255;
    return p;
  };
  const int nbs = (N + 255) / 256;            // 391 <= 512
  int*   degI   = (int*)  carve((size_t)N * 4);
  int*   rowptr = (int*)  carve((size_t)(N + 1) * 4);
  int*   cursor = (int*)  carve((size_t)N * 4);
  int*   bsums  = (int*)  carve((size_t)nbs * 4);
  float* dinv   = (float*)carve((size_t)N * 4);
  float* W2p    = (float*)carve((size_t)Fh * Fcp * 4);
  int*   csrc   = (int*)  carve((size_t)Etot * 4);
  float* Hbuf   = (float*)carve((size_t)N * Fh * 4);   // also holds [N,48] H2
  float* Abuf   = (float*)carve((size_t)N * Fh * 4);

  // degree + CSR (rebuilt every call -> deterministic work)
  k_deg_init <<<(N + 255) / 256, 256, 0, stream>>>(degI, N);
  k_deg_count<<<(E + 255) / 256, 256, 0, stream>>>(dst, E, degI);
  k_scan1    <<<nbs, 256, 0, stream>>>(degI, N, rowptr, bsums);
  k_scan2    <<<1, 512, 0, stream>>>(bsums, nbs);
  k_finish   <<<(N + 255) / 256, 256, 0, stream>>>(rowptr, cursor, bsums, degI, dinv, N, Etot);
  k_fill     <<<(Etot + 255) / 256, 256, 0, stream>>>(src, dst, E, N, cursor, csrc);
  k_pad_w    <<<(Fh * Fcp + 255) / 256, 256, 0, stream>>>(W2, W2p, Fh, Fc, Fcp);

  const int sMt = N / 32;                     // 3125 M-supertiles (MT=2)

  // layer 0: H = x @ W0 ; A0 = relu(agg(H) + b0)
  {
    int waves = sMt * (Fh / 32);              // 12500, /4 = 3125 blocks
    k_gemm<4, Fin, Fin, Fh, 2, 2><<<waves / 4, 128, 0, stream>>>(x, W0, Hbuf);
    k_aggregate<<<N, 128, 0, stream>>>(Hbuf, Fh, dinv, rowptr, csrc, b0, Abuf, Fh, 1);
  }
  // layer 1
  {
    int waves = sMt * (Fh / 32);
    k_gemm<4, Fh, Fh, Fh, 2, 2><<<waves / 4, 128, 0, stream>>>(Abuf, W1, Hbuf);
    k_aggregate<<<N, 128, 0, stream>>>(Hbuf, Fh, dinv, rowptr, csrc, b1, Abuf, Fh, 1);
  }
  // layer 2: H2[N,48] = A1 @ W2p ; out = agg(H2)[:, :40] + b2 (no relu)
  {
    int waves = sMt;                          // NT=3 covers all 48 cols
    k_gemm<1, Fh, Fh, Fcp, 2, 3><<<waves, 32, 0, stream>>>(Abuf, W2p, Hbuf);
    k_aggregate<<<N, 64, 0, stream>>>(Hbuf, Fcp, dinv, rowptr, csrc, b2, (float*)d_out, Fc, 0);
  }
}